// DeformableBlock_1786706395361
// MI455X (gfx1250) — compile-verified
//
#include <hip/hip_runtime.h>

// DeformableBlock for MI455X (gfx1250, wave32, WMMA).
// Pipeline:
//   K1: offset = conv3x3(x, w_off) + b_off          -> d_ws   (WMMA f32 16x16x4 GEMM, M padded 18->32)
//   K2: out    = deform_conv3x3(x, offset, w_dcn)   -> d_out  (bilinear-im2col fused WMMA GEMM, M=64)
// K axis reordered as k' = kk*64 + c so bilinear coords are computed once per tap
// and reused across all 64 channels.

typedef __attribute__((ext_vector_type(2))) float v2f;
typedef __attribute__((ext_vector_type(8))) float v8f;

#define BN    8
#define CIN   64
#define HH    128
#define WW    128
#define KKT   9        // 3x3 taps
#define COFF  18       // 2*K*K offset channels
#define HWSZ  (HH*WW)  // 16384
#define CHW   (CIN*HWSZ)

// ---------------------------------------------------------------------------
// Kernel 1: offset conv.  GEMM: A = w_off reordered [32(M) x 576(K')],
// B = im2col(x) [576 x N], N tiled 16/wave, 128/block (one output row).
// ---------------------------------------------------------------------------
__global__ __launch_bounds__(256) void offset_conv_kernel(
    const float* __restrict__ x, const float* __restrict__ w_off,
    const float* __restrict__ b_off, float* __restrict__ offs)
{
    __shared__ float As[32 * 66];               // per-tap A slice, padded rows
    const int tid  = threadIdx.x;
    const int wave = tid >> 5;
    const int lane = tid & 31;
    const int npix = lane & 15;
    const int half = lane >> 4;

    const int n0 = blockIdx.x * 128;            // one full output row per block
    const int b  = n0 >> 14;                    // n0 / 16384
    const int y  = (n0 & (HWSZ - 1)) >> 7;      // row
    const int xo = (wave << 4) + npix;          // 0..127

    const float* xb = x + b * CHW;

    v8f zero = {0.f,0.f,0.f,0.f,0.f,0.f,0.f,0.f};
    v8f acc0 = zero, acc1 = zero;

    for (int kk = 0; kk < KKT; ++kk) {
        __syncthreads();                        // protect previous slice
        // stage A slice: As[m][c] = w_off[m, c, ky, kx]  (m>=18 -> 0)
        for (int e = tid; e < 32 * 64; e += 256) {
            const int m = e >> 6, c = e & 63;
            As[m * 66 + c] = (m < COFF) ? w_off[m * (CIN * KKT) + c * KKT + kk] : 0.0f;
        }
        __syncthreads();

        const int ky = kk / 3, kx = kk % 3;
        const int iy = y - 1 + ky;
        const int ix = xo - 1 + kx;
        const bool valid = (iy >= 0) && (iy < HH) && (ix >= 0) && (ix < WW);
        const int  idx = (valid ? iy : 0) * WW + (valid ? ix : 0);
        const float vs = valid ? 1.0f : 0.0f;

        for (int c0 = 0; c0 < CIN; c0 += 4) {
            const int c = c0 + 2 * half;        // this lane's channel pair
            v2f bf;
            bf.x = vs * xb[c * HWSZ + idx];
            bf.y = vs * xb[(c + 1) * HWSZ + idx];

            v2f a0, a1;
            a0.x = As[npix * 66 + c];        a0.y = As[npix * 66 + c + 1];
            a1.x = As[(npix + 16) * 66 + c]; a1.y = As[(npix + 16) * 66 + c + 1];

            acc0 = __builtin_amdgcn_wmma_f32_16x16x4_f32(false, a0, false, bf,
                                                         (short)0, acc0, false, false);
            acc1 = __builtin_amdgcn_wmma_f32_16x16x4_f32(false, a1, false, bf,
                                                         (short)0, acc1, false, false);
        }
    }

    // C layout: vgpr r holds (m = r + 8*half [+16*tile], n = lane%16)
    float* ob = offs + b * (COFF * HWSZ) + y * WW + xo;
#pragma unroll
    for (int r = 0; r < 8; ++r) {
        const int ch0 = r + 8 * half;
        ob[ch0 * HWSZ] = acc0[r] + b_off[ch0];
        const int ch1 = 16 + ch0;
        if (ch1 < COFF) ob[ch1 * HWSZ] = acc1[r] + b_off[ch1];
    }
}

// ---------------------------------------------------------------------------
// Kernel 2: deformable conv.  A = w_dcn reordered [64 x 576], B = bilinear
// sampled im2col.  Coords/weights computed once per (tap, pixel), reused
// over 64 channels; 4 independent WMMAs share each B fragment.
// ---------------------------------------------------------------------------
__global__ __launch_bounds__(256) void deform_conv_kernel(
    const float* __restrict__ x, const float* __restrict__ offs,
    const float* __restrict__ w_dcn, float* __restrict__ out)
{
    __shared__ float As[64 * 66];               // 16.5 KB per-tap weight slice
    const int tid  = threadIdx.x;
    const int wave = tid >> 5;
    const int lane = tid & 31;
    const int npix = lane & 15;
    const int half = lane >> 4;

    const int n0 = blockIdx.x * 128;
    const int b  = n0 >> 14;
    const int y  = (n0 & (HWSZ - 1)) >> 7;
    const int xo = (wave << 4) + npix;

    const float* xb = x + b * CHW;

    v8f zero = {0.f,0.f,0.f,0.f,0.f,0.f,0.f,0.f};
    v8f acc[4] = {zero, zero, zero, zero};

    for (int kk = 0; kk < KKT; ++kk) {
        __syncthreads();
        for (int e = tid; e < 64 * 64; e += 256) {
            const int m = e >> 6, c = e & 63;
            As[m * 66 + c] = w_dcn[m * (CIN * KKT) + c * KKT + kk];
        }
        __syncthreads();

        // learned offsets for this (tap, pixel): channels 2*kk (dy), 2*kk+1 (dx)
        const int obase = b * (COFF * HWSZ) + y * WW + xo;
        const float dyo = offs[obase + (2 * kk)     * HWSZ];
        const float dxo = offs[obase + (2 * kk + 1) * HWSZ];

        const float py = (float)(y - 1 + kk / 3) + dyo;
        const float px = (float)(xo - 1 + kk % 3) + dxo;
        const float fy = floorf(py), fx = floorf(px);
        const float wy = py - fy,    wx = px - fx;
        const int y0 = (int)fy, x0 = (int)fx;
        const int y1 = y0 + 1,  x1 = x0 + 1;

        const float vy0 = (y0 >= 0 && y0 < HH) ? 1.0f : 0.0f;
        const float vy1 = (y1 >= 0 && y1 < HH) ? 1.0f : 0.0f;
        const float vx0 = (x0 >= 0 && x0 < WW) ? 1.0f : 0.0f;
        const float vx1 = (x1 >= 0 && x1 < WW) ? 1.0f : 0.0f;

        // fold corner validity into bilinear weights (branch-free inner loop)
        const float w00 = (1.f - wy) * (1.f - wx) * vy0 * vx0;
        const float w01 = (1.f - wy) * wx         * vy0 * vx1;
        const float w10 = wy         * (1.f - wx) * vy1 * vx0;
        const float w11 = wy         * wx         * vy1 * vx1;

        const int cy0 = y0 < 0 ? 0 : (y0 > HH - 1 ? HH - 1 : y0);
        const int cy1 = y1 < 0 ? 0 : (y1 > HH - 1 ? HH - 1 : y1);
        const int cx0 = x0 < 0 ? 0 : (x0 > WW - 1 ? WW - 1 : x0);
        const int cx1 = x1 < 0 ? 0 : (x1 > WW - 1 ? WW - 1 : x1);
        const int i00 = cy0 * WW + cx0, i01 = cy0 * WW + cx1;
        const int i10 = cy1 * WW + cx0, i11 = cy1 * WW + cx1;

        for (int c0 = 0; c0 < CIN; c0 += 4) {
            const int c = c0 + 2 * half;
            const float* p0 = xb + c * HWSZ;
            const float* p1 = p0 + HWSZ;

            v2f bf;
            bf.x = w00 * p0[i00] + w01 * p0[i01] + w10 * p0[i10] + w11 * p0[i11];
            bf.y = w00 * p1[i00] + w01 * p1[i01] + w10 * p1[i10] + w11 * p1[i11];

#pragma unroll
            for (int t = 0; t < 4; ++t) {
                v2f a;
                a.x = As[(npix + 16 * t) * 66 + c];
                a.y = As[(npix + 16 * t) * 66 + c + 1];
                acc[t] = __builtin_amdgcn_wmma_f32_16x16x4_f32(false, a, false, bf,
                                                               (short)0, acc[t], false, false);
            }
        }
    }

    float* ob = out + b * (CIN * HWSZ) + y * WW + xo;
#pragma unroll
    for (int t = 0; t < 4; ++t) {
#pragma unroll
        for (int r = 0; r < 8; ++r) {
            const int m = 16 * t + r + 8 * half;
            ob[m * HWSZ] = acc[t][r];
        }
    }
}

// ---------------------------------------------------------------------------
extern "C" void kernel_launch(void* const* d_in, const int* in_sizes, int n_in,
                              void* d_out, int out_size, void* d_ws, size_t ws_size,
                              hipStream_t stream) {
    const float* x     = (const float*)d_in[0];
    const float* w_off = (const float*)d_in[1];
    const float* b_off = (const float*)d_in[2];
    const float* w_dcn = (const float*)d_in[3];
    float* out  = (float*)d_out;
    float* offs = (float*)d_ws;                 // 8*18*128*128 floats = 9.4 MB

    const int nblocks = BN * HWSZ / 128;        // 1024 blocks, 256 thr (8 waves)
    offset_conv_kernel<<<nblocks, 256, 0, stream>>>(x, w_off, b_off, offs);
    deform_conv_kernel<<<nblocks, 256, 0, stream>>>(x, offs, w_dcn, out);
}